// FlipFlopLoss_13804024889449
// MI455X (gfx1250) — compile-verified
//
#include <hip/hip_runtime.h>
#include <stdint.h>

// Problem constants (match reference: NT, NB, NF, NPOS, SHARP)
constexpr int   NT    = 4000;
constexpr int   NB    = 128;
constexpr int   NF    = 40;
constexpr int   NPOS  = 500;
constexpr float SHARP = 1.0f;

constexpr int BLOCK = 512;   // 16 wave32s; thread p owns DP state p (p < 500)
constexpr int XS    = 4;     // x[t] LDS ring slots (power of two)
constexpr int XPF   = 3;     // async prefetch depth (<= XS-1)

// Per-lane async copy of 4 bytes global -> LDS (tracked by ASYNCcnt).
__device__ __forceinline__ void async_g2l_b32(uint64_t gaddr, uint32_t lds_addr) {
    asm volatile("global_load_async_to_lds_b32 %0, %1, off"
                 :: "v"(lds_addr), "v"(gaddr)
                 : "memory");
}

// Async loads complete in order per wave, so count<=XPF-1 => oldest slot done.
__device__ __forceinline__ void wait_async_steady() {
    asm volatile("s_wait_asynccnt 0x2" ::: "memory");
}
__device__ __forceinline__ void wait_async_all() {
    asm volatile("s_wait_asynccnt 0x0" ::: "memory");
}

// logaddexp via hardware v_exp_f32 / v_log_f32; arg of log is in [1,2].
__device__ __forceinline__ float log_add_exp(float a, float b) {
    float mx = fmaxf(a, b);
    float mn = fminf(a, b);
    return mx + __logf(1.0f + __expf(mn - mx));
}

// One DP timestep: read fwd[t&1], write fwd[(t+1)&1]; gather from xbuf[t&3].
__device__ __forceinline__ void dp_step(int t, int tid, int sidx, int midx,
                                        const float (&xbuf)[XS][NF],
                                        float (&fwd)[2][BLOCK]) {
    const int par = t & 1;
    if (tid < NPOS) {
        const float* xc = xbuf[t & (XS - 1)];
        float s = xc[sidx] * SHARP + fwd[par][tid];
        if (tid > 0) {
            float m = xc[midx] * SHARP + fwd[par][tid - 1];
            s = log_add_exp(m, s);
        }
        fwd[par ^ 1][tid] = s;   // other buffer: no read/write race
    }
}

__global__ __launch_bounds__(BLOCK) void flipflop_fwd_kernel(
    const float* __restrict__ x,         // [NT, NB, NF]
    const int*   __restrict__ move_idx,  // [NB, NPOS-1]
    const int*   __restrict__ stay_idx,  // [NB, NPOS]
    const int*   __restrict__ seqlens,   // [NB]
    float*       __restrict__ out)       // [NB, 1]
{
    __shared__ float xbuf[XS][NF];       // async-filled ring of x[t, b, :]
    __shared__ float fwd[2][BLOCK];      // double-buffered DP state

    const int tid = threadIdx.x;
    const int b   = blockIdx.x;

    // Loop-invariant gather indices, hoisted once for all 4000 steps.
    int sidx = 0, midx = 0;
    if (tid < NPOS) {
        sidx = stay_idx[(size_t)b * NPOS + tid];
        if (tid > 0) midx = move_idx[(size_t)b * (NPOS - 1) + (tid - 1)];
    }

    // fwd0: position 0 = 0, rest = -1e30
    if (tid < NPOS) fwd[0][tid] = (tid == 0) ? 0.0f : -1e30f;

    // Prologue: lanes tid<NF async-load x[0..XPF-1, b, tid] into ring slots.
    uint64_t gaddr = (uint64_t)(uintptr_t)(x + (size_t)b * NF + tid);
    const uint64_t gstep = (uint64_t)NB * NF * sizeof(float);
    if (tid < NF) {
        #pragma unroll
        for (int k = 0; k < XPF; ++k) {
            async_g2l_b32(gaddr, (uint32_t)(uintptr_t)&xbuf[k][tid]);
            gaddr += gstep;
        }
    }

    int t = 0;
    // Steady state: before the wait, loads {t, t+1, t+2} are outstanding;
    // count<=2 (in-order completion) guarantees slot t&3 is filled.
    for (; t + XPF < NT; ++t) {
        wait_async_steady();
        __syncthreads();       // publish slot t&3 (and fwd[t&1]) to all waves

        // Refill slot (t+XPF)&3: it was last read at iteration t-1, and every
        // wave has passed the barrier above since then.
        if (tid < NF) {
            async_g2l_b32(gaddr,
                          (uint32_t)(uintptr_t)&xbuf[(t + XPF) & (XS - 1)][tid]);
            gaddr += gstep;
        }
        dp_step(t, tid, sidx, midx, xbuf, fwd);
    }

    // Drain: fewer than XPF loads may remain outstanding, so a threshold wait
    // no longer proves the current slot is ready -- wait for all of them.
    for (; t < NT; ++t) {
        wait_async_all();
        __syncthreads();
        dp_step(t, tid, sidx, midx, xbuf, fwd);
    }

    __syncthreads();
    if (tid == 0) {
        int sl = seqlens[b];                      // in [1, NPOS]
        out[b] = -fwd[NT & 1][sl - 1] / (SHARP * (float)NT);
    }
}

extern "C" void kernel_launch(void* const* d_in, const int* in_sizes, int n_in,
                              void* d_out, int out_size, void* d_ws, size_t ws_size,
                              hipStream_t stream) {
    (void)in_sizes; (void)n_in; (void)out_size; (void)d_ws; (void)ws_size;
    const float* x        = (const float*)d_in[0];
    const int*   move_idx = (const int*)d_in[1];
    const int*   stay_idx = (const int*)d_in[2];
    const int*   seqlens  = (const int*)d_in[3];
    float*       out      = (float*)d_out;

    flipflop_fwd_kernel<<<dim3(NB), dim3(BLOCK), 0, stream>>>(
        x, move_idx, stay_idx, seqlens, out);
}